// SelfAttention_19250043421053
// MI455X (gfx1250) — compile-verified
//
#include <hip/hip_runtime.h>
#include <hip/hip_bf16.h>

// ---------------------------------------------------------------------------
// SelfAttention with softmax over axis 0 collapses analytically:
//   softmax(.., axis=0) has unit column sums  =>  mean(A@V, 0) == mean(V, 0)
//   out = mean(x,0) @ Wv + bv            (exact, fp32)
// Memory-bound: 32MB (x) + 4MB (Wv) ~= 1.6us at 23.3 TB/s on MI455X.
// All matrix math uses V_WMMA_F32_16X16X4_F32 (full fp32 precision).
// ---------------------------------------------------------------------------

typedef __attribute__((ext_vector_type(2))) float v2f;
typedef __attribute__((ext_vector_type(8))) float v8f;

#define NROW 8192
#define NCOL 1024
#define HDIM 1024
#define ROW_SPLITS 32
#define ROWS_PER 256           // NROW / ROW_SPLITS
#define COL_TILES 64           // NCOL / 16

static __device__ __forceinline__ v8f wmma_f32_4(v2f a, v2f b, v8f c) {
    // D(16x16,f32) = A(16x4,f32) x B(4x16,f32) + C
    return __builtin_amdgcn_wmma_f32_16x16x4_f32(
        /*neg_a=*/false, a, /*neg_b=*/false, b,
        /*c_mod=*/(short)0, c, /*reuse_a=*/false, /*reuse_b=*/false);
}

// Kernel 1: per-(column-tile, row-split) partial column sums of x via WMMA.
// A = all-ones 16x4  =>  D[m,n] = sum_k B[k,n]  (column reduction, exact fp32 FMA tree)
__global__ __launch_bounds__(256) void colsum_wmma(const float* __restrict__ x,
                                                   float* __restrict__ partial) {
    const int lane = threadIdx.x & 31;
    const int wid  = blockIdx.x * 8 + (threadIdx.x >> 5);   // 2048 waves total
    const int ct   = wid & (COL_TILES - 1);                  // column tile [0,64)
    const int rs   = wid >> 6;                               // row split   [0,32)
    const int col0 = ct * 16;
    const int row0 = rs * ROWS_PER;

    const int nsel = lane & 15;           // B: column within tile
    const int ksel = (lane >> 4) << 1;    // B: lanes 0-15 -> K=0/1, lanes 16-31 -> K=2/3

    const float* bp = x + (size_t)(row0 + ksel) * NCOL + col0 + nsel;

    v2f a; a.x = 1.0f; a.y = 1.0f;        // all-ones A matrix
    v8f c0 = {}; v8f c1 = {};

    #pragma unroll 4
    for (int k = 0; k < ROWS_PER; k += 8) {
        v2f b0; b0.x = bp[0];            b0.y = bp[NCOL];
        v2f b1; b1.x = bp[4 * NCOL];     b1.y = bp[5 * NCOL];
        c0 = wmma_f32_4(a, b0, c0);
        c1 = wmma_f32_4(a, b1, c1);
        bp += 8 * NCOL;
    }

    if (lane < 16) {
        // c[0] on lanes 0-15 holds D[M=0, N=lane] == column sum for this slab
        partial[rs * NCOL + col0 + lane] = c0[0] + c1[0];
    }
}

// Kernel 2: deterministic fixed-order reduction over row-splits -> xbar = mean(x,0)
__global__ __launch_bounds__(256) void reduce_partials(const float* __restrict__ partial,
                                                       float* __restrict__ xbar) {
    const int col = blockIdx.x * blockDim.x + threadIdx.x;   // 4 blocks x 256
    float s = 0.0f;
    #pragma unroll
    for (int r = 0; r < ROW_SPLITS; ++r) s += partial[r * NCOL + col];
    xbar[col] = s * (1.0f / (float)NROW);
}

// Kernel 3: out = xbar @ Wv + bv via WMMA (A rows all broadcast xbar chunk)
__global__ __launch_bounds__(256) void matvec_wmma(const float* __restrict__ xbar,
                                                   const float* __restrict__ Wv,
                                                   const float* __restrict__ bv,
                                                   float* __restrict__ out) {
    const int lane = threadIdx.x & 31;
    const int wid  = blockIdx.x * 8 + (threadIdx.x >> 5);    // 64 waves total
    const int h0   = wid * 16;

    const int nsel = lane & 15;
    const int ksel = (lane >> 4) << 1;

    v8f c0 = {}; v8f c1 = {};

    #pragma unroll 2
    for (int d = 0; d < NCOL; d += 8) {
        const int da = d + ksel;
        v2f a0; a0.x = xbar[da];     a0.y = xbar[da + 1];
        v2f a1; a1.x = xbar[da + 4]; a1.y = xbar[da + 5];
        const float* wp = Wv + (size_t)da * HDIM + h0 + nsel;
        v2f b0; b0.x = wp[0];        b0.y = wp[HDIM];
        v2f b1; b1.x = wp[4 * HDIM]; b1.y = wp[5 * HDIM];
        c0 = wmma_f32_4(a0, b0, c0);
        c1 = wmma_f32_4(a1, b1, c1);
    }

    if (lane < 16) {
        out[h0 + lane] = c0[0] + c1[0] + bv[h0 + lane];
    }
}

extern "C" void kernel_launch(void* const* d_in, const int* in_sizes, int n_in,
                              void* d_out, int out_size, void* d_ws, size_t ws_size,
                              hipStream_t stream) {
    (void)in_sizes; (void)n_in; (void)out_size; (void)ws_size;
    const float* x  = (const float*)d_in[0];
    // d_in[1..4] = Wq, bq, Wk, bk : analytically irrelevant (unit column sums of softmax)
    const float* Wv = (const float*)d_in[5];
    const float* bv = (const float*)d_in[6];
    float* out = (float*)d_out;

    float* partial = (float*)d_ws;                    // ROW_SPLITS * NCOL floats (128 KB)
    float* xbar    = partial + ROW_SPLITS * NCOL;     // NCOL floats

    // 2048 waves, 8 per block -> 256 blocks
    colsum_wmma<<<256, 256, 0, stream>>>(x, partial);
    reduce_partials<<<4, 256, 0, stream>>>(partial, xbar);
    // 64 waves, 8 per block -> 8 blocks
    matvec_wmma<<<8, 256, 0, stream>>>(xbar, Wv, bv, out);
}